// Yolov1Loss_34454227648536
// MI455X (gfx1250) — compile-verified
//
#include <hip/hip_runtime.h>
#include <hip/hip_bf16.h>

// ---------------------------------------------------------------------------
// YOLOv1 loss, MI455X (gfx1250, wave32).
// Bandwidth-bound: 192.5 MB read @ 23.3 TB/s => ~8.3 us floor; AI ~0.6 F/B.
// Strategy: ONE thread per (b,h,w) pixel (802816 = 3136 blocks x 256 thr,
// no grid-stride loop); 60 coalesced global_load_b32 per thread (channel
// stride 49 floats -> waves touch contiguous 196 B runs per channel); all
// math in VGPRs; deterministic 2-kernel reduction. Intra-wave reduction uses
// V_WMMA_F32_16X16X4_F32 (A = lane partials, B = ones) -> exact fp32 sum.
// ---------------------------------------------------------------------------

typedef __attribute__((ext_vector_type(2))) float v2f;
typedef __attribute__((ext_vector_type(8))) float v8f;

#define YOLO_C  30
#define YOLO_HW 49
#define EPSF    1e-7f

// Exact fp32 sum of all 32 lane values using one WMMA.
// A (16x4 f32): lanes 0-15 give {K0,K1}, lanes 16-31 give {K2,K3}, M=lane%16.
// We place v in A-VGPR0, 0 in A-VGPR1; B = ones(4x16).
// D[m][n] = sum_k A[m][k] = v_m + v_{m+16}.
// D layout: VGPR r holds M=r (lanes 0-15) / M=r+8 (lanes 16-31).
// Sum of a lane's 8 D regs = half-wave total; xor-16 shuffle adds the halves.
// Requires EXEC == all ones (call only from convergent code).
__device__ __forceinline__ float wave_reduce_sum(float v) {
#if __has_builtin(__builtin_amdgcn_wmma_f32_16x16x4_f32)
    v2f a;  a.x = v;    a.y = 0.0f;
    v2f b;  b.x = 1.0f; b.y = 1.0f;
    v8f c = {};
    c = __builtin_amdgcn_wmma_f32_16x16x4_f32(
            /*neg_a=*/false, a, /*neg_b=*/false, b,
            /*c_mod=*/(short)0, c, /*reuse_a=*/false, /*reuse_b=*/false);
    float s = ((c[0] + c[1]) + (c[2] + c[3])) + ((c[4] + c[5]) + (c[6] + c[7]));
    s += __shfl_xor(s, 16, 32);
    return s;
#else
    for (int m = 16; m >= 1; m >>= 1) v += __shfl_xor(v, m, 32);
    return v;
#endif
}

__device__ __forceinline__ float giou_loss(float gx1, float gy1, float gx2, float gy2,
                                           float px1, float py1, float px2, float py2) {
    float x1 = fmaxf(gx1, px1), y1 = fmaxf(gy1, py1);
    float x2 = fminf(gx2, px2), y2 = fminf(gy2, py2);
    float inter = fmaxf(x2 - x1, 0.0f) * fmaxf(y2 - y1, 0.0f);
    float a1 = (gx2 - gx1) * (gy2 - gy1);
    float a2 = (px2 - px1) * (py2 - py1);
    float uni = a1 + a2 - inter;
    float iou = inter / (uni + EPSF);
    float cx1 = fminf(gx1, px1), cy1 = fminf(gy1, py1);
    float cx2 = fmaxf(gx2, px2), cy2 = fmaxf(gy2, py2);
    float enc = (cx2 - cx1) * (cy2 - cy1);
    float giou = iou - (enc - uni) / (enc + EPSF);
    return 1.0f - giou;
}

__global__ void __launch_bounds__(256)
yolo_loss_partial(const float* __restrict__ gt, const float* __restrict__ pred,
                  float* __restrict__ partial, int npix) {
    const int p = blockIdx.x * blockDim.x + threadIdx.x;

    float acc = 0.0f;
    if (p < npix) {
        int b  = p / YOLO_HW;
        int hw = p - b * YOLO_HW;
        size_t base = (size_t)b * (YOLO_C * YOLO_HW) + hw;

        float g[YOLO_C], q[YOLO_C];
#pragma unroll
        for (int c = 0; c < YOLO_C; ++c) {
            g[c] = gt[base + (size_t)c * YOLO_HW];
            q[c] = pred[base + (size_t)c * YOLO_HW];
        }

        // gt box (chans 0-3) and pred boxes (0-3, 5-8) as xyxy.
        float gx2 = g[0] + g[2], gy2 = g[1] + g[3];
        float l1 = giou_loss(g[0], g[1], gx2, gy2,
                             q[0], q[1], q[0] + q[2], q[1] + q[3]);
        float l2 = giou_loss(g[0], g[1], gx2, gy2,
                             q[5], q[6], q[5] + q[7], q[6] + q[8]);

        // argmax([l1,l2]) with first-occurrence tie-break: l1 >= l2 -> box 0.
        bool  r0  = (l1 >= l2);
        float is0 = r0 ? g[4] : 0.0f;
        float is1 = r0 ? 0.0f : g[9];
        float appears = fmaxf(is0, is1);

        // xy SSE (chans 0,1 masked by is0; chans 5,6 masked by is1), *5.
        float d0 = g[0] - q[0], d1 = g[1] - q[1];
        float d5 = g[5] - q[5], d6 = g[6] - q[6];
        float xy = (d0 * d0 + d1 * d1) * is0 + (d5 * d5 + d6 * d6) * is1;

        // wh SSE on sqrt (chans 2,3 / 7,8), *5.
        float s2 = sqrtf(g[2]) - sqrtf(q[2]), s3 = sqrtf(g[3]) - sqrtf(q[3]);
        float s7 = sqrtf(g[7]) - sqrtf(q[7]), s8 = sqrtf(g[8]) - sqrtf(q[8]);
        float wh = (s2 * s2 + s3 * s3) * is0 + (s7 * s7 + s8 * s8) * is1;

        // confidence: d^2*is + d^2*(1-is)*0.5 for chans 4 and 9.
        float d4 = g[4] - q[4], cd4 = d4 * d4;
        float d9 = g[9] - q[9], cd9 = d9 * d9;
        float conf = cd4 * is0 + cd4 * (1.0f - is0) * 0.5f
                   + cd9 * is1 + cd9 * (1.0f - is1) * 0.5f;

        // class SSE (chans 10..29) * appears.
        float cls = 0.0f;
#pragma unroll
        for (int c = 10; c < YOLO_C; ++c) {
            float d = g[c] - q[c];
            cls += d * d;
        }

        acc = 5.0f * (xy + wh) + conf + appears * cls;
    }

    // Convergent here: intra-wave exact fp32 sum via WMMA, then block combine.
    float wsum = wave_reduce_sum(acc);

    __shared__ float smem[8];
    int lane = threadIdx.x & 31;
    int wave = threadIdx.x >> 5;
    if (lane == 0) smem[wave] = wsum;
    __syncthreads();
    if (threadIdx.x == 0) {
        float s = 0.0f;
        int nw = blockDim.x >> 5;
        for (int w = 0; w < nw; ++w) s += smem[w];   // fixed order -> deterministic
        partial[blockIdx.x] = s;
    }
}

__global__ void __launch_bounds__(256)
yolo_loss_final(const float* __restrict__ partial, int n, float* __restrict__ out) {
    float acc = 0.0f;
    for (int i = threadIdx.x; i < n; i += blockDim.x) acc += partial[i];

    float wsum = wave_reduce_sum(acc);

    __shared__ float smem[8];
    int lane = threadIdx.x & 31;
    int wave = threadIdx.x >> 5;
    if (lane == 0) smem[wave] = wsum;
    __syncthreads();
    if (threadIdx.x == 0) {
        float s = 0.0f;
        int nw = blockDim.x >> 5;
        for (int w = 0; w < nw; ++w) s += smem[w];
        out[0] = s;
    }
}

extern "C" void kernel_launch(void* const* d_in, const int* in_sizes, int n_in,
                              void* d_out, int out_size, void* d_ws, size_t ws_size,
                              hipStream_t stream) {
    const float* gt   = (const float*)d_in[0];
    const float* pred = (const float*)d_in[1];
    float* out        = (float*)d_out;
    float* partial    = (float*)d_ws;          // nblocks * 4 B scratch

    const int npix    = in_sizes[0] / YOLO_C;        // B*H*W = 802816
    const int nblocks = (npix + 255) / 256;          // 3136: one pixel per thread

    yolo_loss_partial<<<nblocks, 256, 0, stream>>>(gt, pred, partial, npix);
    yolo_loss_final<<<1, 256, 0, stream>>>(partial, nblocks, out);
}